// SigModeOv_21852793602398
// MI455X (gfx1250) — compile-verified
//
#include <hip/hip_runtime.h>

#define BB 16
#define LL 2048
#define DD 256

typedef __attribute__((ext_vector_type(2))) float v2f;
typedef __attribute__((ext_vector_type(8))) float v8f;

__device__ __forceinline__ int imin(int a, int b) { return a < b ? a : b; }

// ---------------------------------------------------------------------------
// Kernel 1: s[b,l] = sigmoid(dot(x[b,l,:], W) + bias). One thread per (b,l).
// ---------------------------------------------------------------------------
__global__ void sig_score_kernel(const float* __restrict__ x,
                                 const float* __restrict__ W,
                                 const float* __restrict__ bias,
                                 float* __restrict__ s) {
  int idx = blockIdx.x * blockDim.x + threadIdx.x;  // 0 .. B*L-1
  if (idx >= BB * LL) return;
  const float4* row = (const float4*)(x + (size_t)idx * DD);
  const float4* w4  = (const float4*)W;
  float acc = 0.f;
#pragma unroll 8
  for (int i = 0; i < DD / 4; ++i) {
    float4 xv = row[i];
    float4 wv = w4[i];
    acc += xv.x * wv.x + xv.y * wv.y + xv.z * wv.z + xv.w * wv.w;
  }
  acc += bias[0];
  s[idx] = 1.f / (1.f + expf(-acc));
}

// ---------------------------------------------------------------------------
// Kernel 2: valley detection + ordered compaction of segment starts/ends.
// One wave (32 lanes) per batch row; ballot+popcount keeps ascending order.
//   start list = {0} ∪ valleys ; end list = {v+2 for valleys} ∪ {L}
// Also writes output_lengths[b] = int((ilen[b]/ilen[0]) * M).
// ---------------------------------------------------------------------------
__global__ void valley_kernel(const float* __restrict__ s,
                              const int* __restrict__ ilen,
                              int* __restrict__ starts,
                              int* __restrict__ ends,
                              int* __restrict__ counts,
                              int* __restrict__ out_len,
                              int Mv) {
  int b = blockIdx.x;
  int lane = threadIdx.x & 31;
  const float* srow = s + (size_t)b * LL;
  int nstart = 0, nend = 0;
  unsigned lanemask = (1u << lane) - 1u;
  for (int base = 0; base < LL; base += 32) {
    int l = base + lane;
    float cur = srow[l];
    float before = (l == 0) ? 0.f : srow[l - 1];
    float after  = (l == LL - 1) ? 0.f : srow[l + 1];
    bool valley = (cur < before) && (cur < after);
    bool is_start = valley || (l == 0);
    bool is_end   = valley || (l == LL - 1);
    unsigned ms = (unsigned)__ballot(is_start);
    unsigned me = (unsigned)__ballot(is_end);
    if (is_start) {
      int idx = nstart + __popc(ms & lanemask);
      if (idx < Mv) starts[b * Mv + idx] = l;
    }
    if (is_end) {
      int idx = nend + __popc(me & lanemask);
      if (idx < Mv) ends[b * Mv + idx] = imin(l + 2, LL);
    }
    nstart += __popc(ms);
    nend += __popc(me);
  }
  if (lane == 0) {
    counts[b] = imin(nstart, Mv);
    float ratio = (float)ilen[b] / (float)ilen[0];
    out_len[b] = (int)(ratio * (float)Mv);
  }
}

// ---------------------------------------------------------------------------
// Kernel 3: banded masked matmul on WMMA f32 16x16x4.
//   A[m,k] = (start_m <= k < end_m) ? s[k] : 0   (16 segments x 4 frames)
//   B[k,n] = x[b, k, d0+n]
//   C      = 16x16 tile of num;   den[m] = row-sum of A (via lane partials)
//   out    = C / max(den, 1e-6)
//
// Fully branch-free inner loop: all loads use clamped (always in-bounds)
// indices; the s-loads are pinned with an empty asm "+v" use so LLVM cannot
// sink them into a predicated block, and masking is pure v_cndmask. EXEC is
// all-ones at every instruction. Overshoot past `hi` is benign: A is
// identically zero there, so the extra B values are multiplied by 0.
// ---------------------------------------------------------------------------
__device__ __forceinline__ void wmma_step(int l, int half, int st, int en,
                                          const float* __restrict__ srow,
                                          const float* __restrict__ xcol,
                                          float& denp, v8f& c) {
  int k0 = l + 2 * half;           // A/B reg0 K index for this lane
  int k1 = k0 + 1;                 // A/B reg1 K index
  int k0c = imin(k0, LL - 1);      // clamped, always in-bounds
  int k1c = imin(k1, LL - 1);

  float s0 = srow[k0c];
  float s1 = srow[k1c];
  float b0 = xcol[(size_t)k0c * DD];
  float b1 = xcol[(size_t)k1c * DD];
  // Pin loads as unconditional: prevents sinking into exec-masked blocks.
  asm("" : "+v"(s0), "+v"(s1), "+v"(b0), "+v"(b1));

  float a0 = (k0 >= st && k0 < en) ? s0 : 0.f;   // v_cndmask, no branch
  float a1 = (k1 >= st && k1 < en) ? s1 : 0.f;
  denp += a0 + a1;

  v2f a, bm;
  a[0] = a0; a[1] = a1;
  bm[0] = b0; bm[1] = b1;
  c = __builtin_amdgcn_wmma_f32_16x16x4_f32(
      /*neg_a=*/false, a, /*neg_b=*/false, bm,
      /*c_mod=*/(short)0, c, /*reuse_a=*/false, /*reuse_b=*/false);
}

__global__ void seg_wmma_kernel(const float* __restrict__ x,
                                const float* __restrict__ s,
                                const int* __restrict__ starts,
                                const int* __restrict__ ends,
                                const int* __restrict__ counts,
                                float* __restrict__ out,
                                int Mv) {
  int d0 = blockIdx.x * 16;
  int m0 = blockIdx.y * 16;
  int b  = blockIdx.z;
  int lane = threadIdx.x & 31;
  int half = lane >> 4;      // 0: lanes 0-15, 1: lanes 16-31
  int ml = lane & 15;        // A row index within tile / B column index

  int cnt = counts[b];

  // Per-lane segment bounds for A row (m0+ml); empty if invalid.
  int st = 0, en = 0;
  int mseg = m0 + ml;
  if (mseg < cnt && mseg < Mv) {
    st = starts[b * Mv + mseg];
    en = ends[b * Mv + mseg];
  }

  // Uniform band covered by this 16-segment tile (starts/ends ascending).
  int lo = 0, hi = 0;
  int lastm = imin(imin(m0 + 15, Mv - 1), cnt - 1);
  if (m0 < cnt && m0 < Mv) {
    lo = starts[b * Mv + m0];
    hi = ends[b * Mv + lastm];
  }

  v8f c = {};
  float denp = 0.f;  // partial row-sum of A held by this lane

  const float* srow = s + (size_t)b * LL;
  const float* xcol = x + (size_t)b * LL * DD + d0 + ml;  // column d0+ml

  // 8 K-frames (2 WMMAs) per trip; overshoot past hi is masked to zero by A.
  for (int l = lo; l < hi; l += 8) {
    wmma_step(l,     half, st, en, srow, xcol, denp, c);
    wmma_step(l + 4, half, st, en, srow, xcol, denp, c);
  }

  // Combine K-half partials: lanes m and m+16 hold the two halves of den[m].
  float dtot = denp + __shfl_xor(denp, 16, 32);

  __shared__ float den_sh[16];
  if (lane < 16) den_sh[lane] = dtot;
  __syncthreads();

  // C layout: reg r -> row r + 8*half, column ml.
#pragma unroll
  for (int r = 0; r < 8; ++r) {
    int row = r + 8 * half;
    int m = m0 + row;
    if (m < Mv) {
      float den = fmaxf(den_sh[row], 1e-6f);
      out[(size_t)b * Mv * DD + (size_t)m * DD + d0 + ml] = c[r] / den;
    }
  }
}

extern "C" void kernel_launch(void* const* d_in, const int* in_sizes, int n_in,
                              void* d_out, int out_size, void* d_ws, size_t ws_size,
                              hipStream_t stream) {
  const float* x    = (const float*)d_in[0];  // [B,L,D]
  const float* W    = (const float*)d_in[1];  // [D,1]
  const float* bias = (const float*)d_in[2];  // [1]
  const int*   ilen = (const int*)d_in[3];    // [B]

  // out = concat(output[B,M,D], output_lengths[B] (int bits), s[B,L,1])
  int Mv = (out_size - BB - BB * LL) / (BB * DD);

  float* out_main = (float*)d_out;
  int*   out_len  = (int*)((float*)d_out + (size_t)BB * Mv * DD);
  float* out_s    = (float*)d_out + (size_t)BB * Mv * DD + BB;

  // Workspace: starts[B*M], ends[B*M], counts[B]
  int* starts = (int*)d_ws;
  int* ends   = starts + (size_t)BB * Mv;
  int* counts = ends + (size_t)BB * Mv;

  // Stage 1: scores
  sig_score_kernel<<<(BB * LL + 255) / 256, 256, 0, stream>>>(x, W, bias, out_s);

  // Stage 2: segment lists + output_lengths
  valley_kernel<<<BB, 32, 0, stream>>>(out_s, ilen, starts, ends, counts,
                                       out_len, Mv);

  // Stage 3: banded WMMA matmul + normalize
  dim3 grid(DD / 16, (Mv + 15) / 16, BB);
  seg_wmma_kernel<<<grid, 32, 0, stream>>>(x, out_s, starts, ends, counts,
                                           out_main, Mv);
}